// AdditiveAttender_58617713656236
// MI455X (gfx1250) — compile-verified
//
#include <hip/hip_runtime.h>
#include <hip/hip_bf16.h>

// Additive (Bahdanau) attention, fused:
//   hkT[b,d,k] = (keys[b]@Wk + b1)[k,d]   (WMMA f32 GEMM, transposed store)
//   hq [b,q,d] = (queries[b]@Wq)[q,d]     (WMMA f32 GEMM)
//   attn[b,q,k] = softmax_k( sum_d w2[d]*tanh(hq+hkT) + b2 )   (VALU/TRANS: v_tanh_f32)
//   out[b,q,:] = attn[b]@values[b]        (WMMA f32 GEMM)

constexpr int kB   = 16;
constexpr int kNQ  = 256;
constexpr int kNK  = 256;
constexpr int kD   = 256;
constexpr int kQPB = 4;    // queries per block in the score kernel

typedef float v2f __attribute__((ext_vector_type(2)));
typedef float v8f __attribute__((ext_vector_type(8)));

#if __has_builtin(__builtin_amdgcn_wmma_f32_16x16x4_f32)
#define HAVE_WMMA_F32X4 1
#else
#define HAVE_WMMA_F32X4 0
#endif

__device__ __forceinline__ float fast_exp2(float x) {
#if __has_builtin(__builtin_amdgcn_exp2f)
    return __builtin_amdgcn_exp2f(x);
#else
    return exp2f(x);
#endif
}

__device__ __forceinline__ float fast_tanh(float x) {
#if __has_builtin(__builtin_amdgcn_tanhf)
    return __builtin_amdgcn_tanhf(x);          // v_tanh_f32 (gfx1250)
#else
    // tanh(x) = 1 - 2/(exp(2x)+1), exp via v_exp_f32
    float e = fast_exp2(x * 2.8853900817779268f);  // 2*log2(e)
    return 1.0f - 2.0f / (e + 1.0f);
#endif
}

// One wave (32 threads) per 16(M) x 32(N) output tile; K-loop in steps of 4.
// A: [M,K] row-major (per batch), Bm: [K,N] row-major, O: [M,N] (or [N,M] if transposeO).
__global__ void wmma_gemm_f32(const float* __restrict__ A, long long sA,
                              const float* __restrict__ Bm, long long sB,
                              const float* __restrict__ bias,   // len N or nullptr
                              float* __restrict__ O, long long sO,
                              int M, int N, int K, int transposeO)
{
    const float* Ab = A + (long long)blockIdx.z * sA;
    const float* Bb = Bm + (long long)blockIdx.z * sB;
    float*       Ob = O + (long long)blockIdx.z * sO;

    const int tm   = blockIdx.y * 16;
    const int tn   = blockIdx.x * 32;
    const int lane = (int)(threadIdx.x & 31u);
    const int half = lane >> 4;    // selects K pair (A) / K pair (B)
    const int l16  = lane & 15;

    float acc0[8], acc1[8];

#if HAVE_WMMA_F32X4
    v8f c0 = {0.f,0.f,0.f,0.f,0.f,0.f,0.f,0.f};
    v8f c1 = {0.f,0.f,0.f,0.f,0.f,0.f,0.f,0.f};
    const float* arow = Ab + (long long)(tm + l16) * K;
#pragma unroll 4
    for (int k0 = 0; k0 < K; k0 += 4) {
        const int ka = k0 + half * 2;
        // A 16x4 frag: lanes 0-15 rows M, VGPR0=K(ka), VGPR1=K(ka+1); lanes 16-31 -> K+2
        v2f a  = { arow[ka], arow[ka + 1] };
        const float* brow = Bb + (long long)ka * N;
        // B 4x16 frag: lanes hold column n=l16; VGPR0=row ka, VGPR1=row ka+1; half selects K+2
        v2f b0 = { brow[tn + l16],       brow[N + tn + l16] };
        v2f b1 = { brow[tn + 16 + l16],  brow[N + tn + 16 + l16] };
        c0 = __builtin_amdgcn_wmma_f32_16x16x4_f32(false, a, false, b0, (short)0, c0, false, false);
        c1 = __builtin_amdgcn_wmma_f32_16x16x4_f32(false, a, false, b1, (short)0, c1, false, false);
    }
#pragma unroll
    for (int r = 0; r < 8; ++r) { acc0[r] = c0[r]; acc1[r] = c1[r]; }
#else
    // Scalar fallback with identical C layout (compile-safety net).
#pragma unroll
    for (int r = 0; r < 8; ++r) {
        const int m = tm + r + half * 8;
        float s0 = 0.f, s1 = 0.f;
        for (int k = 0; k < K; ++k) {
            const float av = Ab[(long long)m * K + k];
            s0 = fmaf(av, Bb[(long long)k * N + tn + l16], s0);
            s1 = fmaf(av, Bb[(long long)k * N + tn + 16 + l16], s1);
        }
        acc0[r] = s0; acc1[r] = s1;
    }
#endif

    // C/D layout: VGPR r, lanes 0-15 -> M=r, lanes 16-31 -> M=r+8; N = lane&15
#pragma unroll
    for (int r = 0; r < 8; ++r) {
        const int m  = tm + r + half * 8;
        const int n0 = tn + l16;
        const int n1 = tn + 16 + l16;
        float v0 = acc0[r] + (bias ? bias[n0] : 0.f);
        float v1 = acc1[r] + (bias ? bias[n1] : 0.f);
        if (transposeO) {
            Ob[(long long)n0 * M + m] = v0;
            Ob[(long long)n1 * M + m] = v1;
        } else {
            Ob[(long long)m * N + n0] = v0;
            Ob[(long long)m * N + n1] = v1;
        }
    }
}

// One block (256 threads) per (b, 4 queries). Thread t owns key index k=t.
// Each hkT element loaded once per block feeds kQPB tanh+fma ops (AI x4 vs naive).
__global__ void additive_scores_softmax(const float* __restrict__ hq,   // [B,NQ,D]
                                        const float* __restrict__ hkT,  // [B,D,NK] (bias folded)
                                        const float* __restrict__ w2,   // [D]
                                        const float* __restrict__ b2,   // [1]
                                        float* __restrict__ attn)       // [B,NQ,NK]
{
    const int blk = (int)blockIdx.x;            // 0 .. B*NQ/kQPB - 1
    const int b   = blk / (kNQ / kQPB);
    const int q0  = (blk % (kNQ / kQPB)) * kQPB;
    const int t   = (int)threadIdx.x;           // 0..255 (key index)

    __shared__ float s_hq[kQPB][kD];
    __shared__ float s_w2[kD];
    __shared__ float s_red[kNK];

    s_w2[t] = w2[t];
#pragma unroll
    for (int i = 0; i < kQPB; ++i)
        s_hq[i][t] = hq[((long long)b * kNQ + q0 + i) * kD + t];
    __syncthreads();

    const float* hkTb = hkT + (long long)b * kD * kNK;
    float acc[kQPB] = {0.f, 0.f, 0.f, 0.f};
#pragma unroll 2
    for (int d = 0; d < kD; ++d) {
        const float hk = hkTb[(long long)d * kNK + t];   // coalesced over t
        const float w  = s_w2[d];                        // LDS broadcast
#pragma unroll
        for (int i = 0; i < kQPB; ++i)
            acc[i] = fmaf(w, fast_tanh(s_hq[i][d] + hk), acc[i]);
    }
    const float b2v = b2[0];

    for (int i = 0; i < kQPB; ++i) {
        const float logit = acc[i] + b2v;

        // max-reduce over 256 threads
        s_red[t] = logit;
        __syncthreads();
#pragma unroll
        for (int s = kNK / 2; s > 0; s >>= 1) {
            if (t < s) s_red[t] = fmaxf(s_red[t], s_red[t + s]);
            __syncthreads();
        }
        const float mx = s_red[0];
        __syncthreads();

        const float p = fast_exp2((logit - mx) * 1.44269504088896f);  // v_exp_f32
        s_red[t] = p;
        __syncthreads();
#pragma unroll
        for (int s = kNK / 2; s > 0; s >>= 1) {
            if (t < s) s_red[t] += s_red[t + s];
            __syncthreads();
        }
        const float inv = 1.0f / s_red[0];
        __syncthreads();   // protect s_red before next query's reduction

        attn[((long long)b * kNQ + q0 + i) * kNK + t] = p * inv;
    }
}

extern "C" void kernel_launch(void* const* d_in, const int* in_sizes, int n_in,
                              void* d_out, int out_size, void* d_ws, size_t ws_size,
                              hipStream_t stream) {
    (void)in_sizes; (void)n_in; (void)out_size; (void)ws_size;

    const float* keys    = (const float*)d_in[0];  // [B,NK,D]
    const float* queries = (const float*)d_in[1];  // [B,NQ,D]
    const float* values  = (const float*)d_in[2];  // [B,NK,D]
    const float* Wk      = (const float*)d_in[3];  // [D,D]
    const float* Wq      = (const float*)d_in[4];  // [D,D]
    const float* b1      = (const float*)d_in[5];  // [D]
    const float* w2      = (const float*)d_in[6];  // [D]
    const float* b2      = (const float*)d_in[7];  // [1]
    float*       out     = (float*)d_out;          // [B,NQ,D]

    // Workspace carve-up (all L2-resident: 12 MB total)
    float* hkT  = (float*)d_ws;                                  // [B,D,NK]  4 MB
    float* hq   = hkT + (size_t)kB * kD * kNK;                   // [B,NQ,D]  4 MB
    float* attn = hq  + (size_t)kB * kNQ * kD;                   // [B,NQ,NK] 4 MB

    const dim3 blk(32);

    // hkT = (keys @ Wk + b1)^T per batch
    wmma_gemm_f32<<<dim3(kD / 32, kNK / 16, kB), blk, 0, stream>>>(
        keys, (long long)kNK * kD, Wk, 0, b1,
        hkT, (long long)kD * kNK, kNK, kD, kD, /*transposeO=*/1);

    // hq = queries @ Wq per batch
    wmma_gemm_f32<<<dim3(kD / 32, kNQ / 16, kB), blk, 0, stream>>>(
        queries, (long long)kNQ * kD, Wq, 0, nullptr,
        hq, (long long)kNQ * kD, kNQ, kD, kD, /*transposeO=*/0);

    // fused tanh-score + softmax (4 queries per block)
    additive_scores_softmax<<<dim3(kB * kNQ / kQPB), dim3(kNK), 0, stream>>>(
        hq, hkT, w2, b2, attn);

    // context = attn @ values per batch
    wmma_gemm_f32<<<dim3(kD / 32, kNQ / 16, kB), blk, 0, stream>>>(
        attn, (long long)kNQ * kNK, values, (long long)kNK * kD, nullptr,
        out, (long long)kNQ * kD, kNQ, kD, kNK, /*transposeO=*/0);
}